// SSDDecoder_53240414601571
// MI455X (gfx1250) — compile-verified
//
#include <hip/hip_runtime.h>
#include <hip/hip_fp16.h>

// ---------------- problem constants ----------------
#define BB 16
#define NN 24564
#define CC 81
#define KK 50            // MAX_PER_CLASS
#define TT 200           // MAX_TOTAL
#define MM (CC * KK)     // 4050 candidates per batch
#define MPAD 4064        // 16*254, multiple of 4 and 16
#define NPAD 24576       // NN rounded up
#define NMS_THREADS 512
#define IOU_THR 0.5f
#define SCORE_THR 0.5f
#define EPSV 1e-8f

typedef float v8f  __attribute__((ext_vector_type(8)));
typedef float v2f  __attribute__((ext_vector_type(2)));
typedef unsigned int u32x4 __attribute__((ext_vector_type(4)));
typedef int   i32x8 __attribute__((ext_vector_type(8)));
typedef int   i32x4 __attribute__((ext_vector_type(4)));

// ---------------- TDM helper: 1D row (nelem x 4B) global -> LDS ----------------
// Builds a D# per CDNA5 ISA: group0 = {ctrl, lds_addr,
// gaddr_lo, gaddr_hi|type=2}; group1 carries data_size=4B, tensor_dim0 = nelem,
// tensor_dim1 = 1, tile_dim0 = nelem (<= 65535), tile_dim1 = 1, stride0 = nelem.
// Groups 2/3 zero (dims <= 2D). clang-23 6-arg builtin form:
// (u32x4, i32x8, i32x4, i32x4, i32x8, i32 cpol).
__device__ __forceinline__ void tdm_load_row_b32(void* lds_dst, const void* gsrc,
                                                 unsigned nelem) {
  unsigned lds = (unsigned)(unsigned long long)lds_dst;           // low 32 = LDS offset
  unsigned long long ga = (unsigned long long)gsrc;
  u32x4 g0;
  g0[0] = 1u;                                                     // count=1 (valid)
  g0[1] = lds;                                                    // lds_addr
  g0[2] = (unsigned)ga;                                           // global_addr[31:0]
  g0[3] = (unsigned)((ga >> 32) & 0x1ffffffull) | (2u << 30);     // gaddr hi | type=2
  i32x8 g1;
  g1[0] = (int)(2u << 16);                                        // data_size=4B
  g1[1] = (int)((nelem & 0xffffu) << 16);                         // tensor_dim0 lo16
  g1[2] = (int)((nelem >> 16) | (1u << 16));                      // td0 hi16 | td1 lo16=1
  g1[3] = (int)((nelem & 0xffffu) << 16);                         // td1 hi16=0 | tile_dim0
  g1[4] = 1;                                                      // tile_dim1=1, tile_dim2=0
  g1[5] = (int)nelem;                                             // tensor_dim0_stride lo32
  g1[6] = 0;
  g1[7] = 0;
  i32x4 z4; z4[0] = z4[1] = z4[2] = z4[3] = 0;
  i32x8 z8; z8[0] = z8[1] = z8[2] = z8[3] = z8[4] = z8[5] = z8[6] = z8[7] = 0;
  __builtin_amdgcn_tensor_load_to_lds(g0, g1, z4, z4, z8, 0);
}

// ---------------- kernel 1: decode + background mask + transpose + pack ----------------
__global__ __launch_bounds__(256) void k_decode_mask(
    const float* __restrict__ deltas, const float* __restrict__ labels,
    const float* __restrict__ priors, float* __restrict__ scoresT,
    float4* __restrict__ boxesF, __half2* __restrict__ bh1,
    __half2* __restrict__ bh2) {
  int gid = blockIdx.x * 256 + threadIdx.x;
  if (gid >= BB * NN) return;
  int b = gid / NN;
  int n = gid - b * NN;

  float4 pr = reinterpret_cast<const float4*>(priors)[n];
  float ph = pr.z - pr.x, pw = pr.w - pr.y;
  float pcy = pr.x + 0.5f * ph, pcx = pr.y + 0.5f * pw;

  float4 dl = reinterpret_cast<const float4*>(deltas)[gid];
  float cy = dl.x * 0.1f * ph + pcy;         // variances fused
  float cx = dl.y * 0.1f * pw + pcx;
  float h  = __expf(dl.z * 0.2f) * ph;
  float w  = __expf(dl.w * 0.2f) * pw;
  float y1 = fminf(fmaxf(cy - 0.5f * h, 0.f), 1.f);
  float x1 = fminf(fmaxf(cx - 0.5f * w, 0.f), 1.f);
  float y2 = fminf(fmaxf(cy + 0.5f * h, 0.f), 1.f);
  float x2 = fminf(fmaxf(cx + 0.5f * w, 0.f), 1.f);

  boxesF[gid] = make_float4(y1, x1, y2, x2);          // full-precision, for output gather
  bh1[gid] = __floats2half2_rn(y1, x1);               // fp16 pair for LDS-resident IoU
  bh2[gid] = __floats2half2_rn(y2, x2);

  // argmax over classes (first occurrence, like jnp.argmax)
  const float* lr = labels + (size_t)gid * CC;
  float mx = lr[0];
  int am = 0;
#pragma unroll 1
  for (int c = 1; c < CC; ++c) {
    float v = lr[c];
    if (v > mx) { mx = v; am = c; }
  }
  float keep = (am != 0) ? 1.f : 0.f;
  // transposed masked scores: (B,C,N) -> lanes write consecutive n => coalesced per c
  float* st = scoresT + (size_t)b * CC * NN + n;
#pragma unroll 1
  for (int c = 0; c < CC; ++c) st[(size_t)c * NN] = lr[c] * keep;
}

// ---------------- kernel 2: per-(b,c) greedy NMS, fully LDS-resident ----------------
__global__ __launch_bounds__(NMS_THREADS, 1) void k_nms(
    const float* __restrict__ scoresT, const __half2* __restrict__ bh1,
    const __half2* __restrict__ bh2, float* __restrict__ candS,
    int* __restrict__ candI) {
  __shared__ float   s_sc[NPAD];    //  96 KB mutable class scores
  __shared__ __half2 s_b1[NPAD];    //  96 KB (y1,x1) fp16
  __shared__ __half2 s_b2[NPAD];    //  96 KB (y2,x2) fp16
  __shared__ float r_s[NMS_THREADS / 32];
  __shared__ int   r_i[NMS_THREADS / 32];
  __shared__ float cur_s;
  __shared__ int   cur_j;

  const int bc = blockIdx.x;        // = b*CC + c, matches scoresT row layout
  const int b  = bc / CC;
  const int tid  = threadIdx.x;
  const int lane = tid & 31;
  const int wid  = tid >> 5;

  // TDM DMA staging: wave 0 issues 3 descriptors (EXEC-independent), waits, barrier.
  if (tid < 32) {
    tdm_load_row_b32(s_sc, scoresT + (size_t)bc * NN, NN);
    tdm_load_row_b32(s_b1, bh1 + (size_t)b * NN, NN);
    tdm_load_row_b32(s_b2, bh2 + (size_t)b * NN, NN);
    __builtin_amdgcn_s_wait_tensorcnt(0);
  }
  __syncthreads();

  float bs;
  int bi;
#define NMS_REDUCE()                                                      \
  do {                                                                    \
    for (int off = 16; off > 0; off >>= 1) {                              \
      float os = __shfl_xor(bs, off, 32);                                 \
      int   oi = __shfl_xor(bi, off, 32);                                 \
      if (os > bs || (os == bs && oi < bi)) { bs = os; bi = oi; }         \
    }                                                                     \
    if (lane == 0) { r_s[wid] = bs; r_i[wid] = bi; }                      \
    __syncthreads();                                                      \
    if (tid == 0) {                                                       \
      float cs = r_s[0]; int cj = r_i[0];                                 \
      for (int wv = 1; wv < NMS_THREADS / 32; ++wv) {                     \
        float os = r_s[wv]; int oi = r_i[wv];                             \
        if (os > cs || (os == cs && oi < cj)) { cs = os; cj = oi; }       \
      }                                                                   \
      cur_s = cs; cur_j = cj;                                             \
    }                                                                     \
    __syncthreads();                                                      \
  } while (0)

  // pass 0: pure argmax (includes zeros -> all-zero row selects index 0 like jnp.argmax)
  bs = -1.f; bi = 0x7fffffff;
  for (int i = tid; i < NN; i += NMS_THREADS) {
    float sc = s_sc[i];
    if (sc > bs || (sc == bs && i < bi)) { bs = sc; bi = i; }
  }
  NMS_REDUCE();

  for (int it = 0; it < KK; ++it) {
    float s = cur_s;
    int   j = cur_j;
    if (tid == 0) {  // record BEFORE suppression (matches reference body order)
      candS[(size_t)bc * KK + it] = (s > SCORE_THR) ? s : 0.f;
      candI[(size_t)bc * KK + it] = j;
    }
    __half2 h1 = s_b1[j], h2 = s_b2[j];  // broadcast LDS read
    float by1 = __low2float(h1), bx1 = __high2float(h1);
    float by2 = __low2float(h2), bx2 = __high2float(h2);
    float a0 = (by2 - by1) * (bx2 - bx1);

    // fused suppress + next-argmax sweep
    bs = -1.f; bi = 0x7fffffff;
    for (int i = tid; i < NN; i += NMS_THREADS) {
      float sc = s_sc[i];
      if (i == j) {                         // explicit .at[j].set(0) semantics
        if (sc != 0.f) { sc = 0.f; s_sc[i] = 0.f; }
      } else if (sc > 0.f) {
        __half2 q1 = s_b1[i], q2 = s_b2[i];
        float y1 = __low2float(q1), x1 = __high2float(q1);
        float y2 = __low2float(q2), x2 = __high2float(q2);
        float ih = fminf(by2, y2) - fmaxf(by1, y1);
        float iw = fminf(bx2, x2) - fmaxf(bx1, x1);
        float inter = fmaxf(ih, 0.f) * fmaxf(iw, 0.f);
        float a1 = (y2 - y1) * (x2 - x1);
        float iou = inter / (a0 + a1 - inter + EPSV);
        if (iou > IOU_THR) { sc = 0.f; s_sc[i] = 0.f; }
      }
      if (sc > bs || (sc == bs && i < bi)) { bs = sc; bi = i; }
    }
    NMS_REDUCE();
  }
#undef NMS_REDUCE
}

// ---------------- kernel 3: WMMA rank-based top-200 + gather ----------------
// rank(p) = #{q : s_q > s_p  or (s_q == s_p and q < p)}  -> stable descending order.
// Computed as (16 x 4 comparison tile) x (4 x 16 ones) WMMA accumulation: every
// column of D holds the row sums, i.e. the ranks. rank < 200 => output slot = rank.
__global__ __launch_bounds__(256) void k_topk(
    const float* __restrict__ candS, const int* __restrict__ candI,
    const float4* __restrict__ boxesF, float* __restrict__ outB,
    float* __restrict__ outS, float* __restrict__ outC,
    float* __restrict__ outN) {
  __shared__ float s_cs[MPAD];
  __shared__ int s_cnt;
  const int b = blockIdx.x;
  const int tid = threadIdx.x;
  for (int i = tid; i < MPAD; i += 256)
    s_cs[i] = (i < MM) ? candS[(size_t)b * MM + i] : -1.f;  // pad < all real scores (>=0)
  if (tid == 0) s_cnt = 0;
  __syncthreads();

  const int lane = tid & 31;
  const int wave = tid >> 5;
  const int m  = lane & 15;   // A-matrix row this lane holds (both halves)
  const int kh = lane >> 4;   // lanes 16-31 hold K=2,3
  v2f ones; ones.x = 1.f; ones.y = 1.f;

  for (int tile = wave; tile < MPAD / 16; tile += 8) {   // uniform per wave: EXEC all-1s
    const int p = tile * 16 + m;
    const float sp = s_cs[p];
    v8f acc = {0.f, 0.f, 0.f, 0.f, 0.f, 0.f, 0.f, 0.f};
    for (int kc = 0; kc < MPAD; kc += 4) {
      int q0 = kc + 2 * kh;
      int q1 = q0 + 1;
      float sq0 = s_cs[q0];   // broadcast reads, no bank conflicts
      float sq1 = s_cs[q1];
      v2f a;
      a.x = ((sq0 > sp) || (sq0 == sp && q0 < p)) ? 1.f : 0.f;
      a.y = ((sq1 > sp) || (sq1 == sp && q1 < p)) ? 1.f : 0.f;
      acc = __builtin_amdgcn_wmma_f32_16x16x4_f32(false, a, false, ones,
                                                  (short)0, acc, false, false);
    }
    // D layout: rows 0-7 in VGPR r lanes 0-15; rows 8-15 in VGPR r-8 lanes 16-31.
    float rk = 0.f;
#pragma unroll
    for (int mm = 0; mm < 16; ++mm) {
      float v = __shfl(acc[(mm < 8) ? mm : (mm - 8)], (mm < 8) ? 0 : 16, 32);
      if (lane == mm) rk = v;
    }
    if (lane < 16 && p < MM) {
      int rank = (int)(rk + 0.5f);   // exact small integer in f32
      if (rank < TT) {
        float s = sp;
        float vm = (s > 0.f) ? 1.f : 0.f;
        int cls = p / KK;
        int bi2 = candI[(size_t)b * MM + p];
        float4 bx = boxesF[(size_t)b * NN + bi2];
        int o = b * TT + rank;
        reinterpret_cast<float4*>(outB)[o] =
            make_float4(bx.x * vm, bx.y * vm, bx.z * vm, bx.w * vm);
        outS[o] = s;
        outC[o] = (float)cls * vm;
        if (vm > 0.f) atomicAdd(&s_cnt, 1);
      }
    }
  }
  __syncthreads();
  if (tid == 0) outN[b] = (float)s_cnt;
}

// ---------------- launch ----------------
extern "C" void kernel_launch(void* const* d_in, const int* in_sizes, int n_in,
                              void* d_out, int out_size, void* d_ws, size_t ws_size,
                              hipStream_t stream) {
  (void)in_sizes; (void)n_in; (void)out_size; (void)ws_size;
  const float* deltas = (const float*)d_in[0];   // (B,N,4)
  const float* labels = (const float*)d_in[1];   // (B,N,C)
  const float* priors = (const float*)d_in[2];   // (N,4)

  char* ws = (char*)d_ws;
  size_t off = 0;
  auto take = [&](size_t bytes) -> void* {
    void* p = ws + off;
    off += (bytes + 255) & ~(size_t)255;
    return p;
  };
  float*   scoresT = (float*)take((size_t)BB * CC * NN * 4);  // ~127 MB
  float4*  boxesF  = (float4*)take((size_t)BB * NN * 16);     // ~6.3 MB
  __half2* bh1     = (__half2*)take((size_t)BB * NN * 4);
  __half2* bh2     = (__half2*)take((size_t)BB * NN * 4);
  float*   candS   = (float*)take((size_t)BB * MM * 4);
  int*     candI   = (int*)take((size_t)BB * MM * 4);

  float* outB = (float*)d_out;                 // (B,200,4)
  float* outS = outB + (size_t)BB * TT * 4;    // (B,200)
  float* outC = outS + (size_t)BB * TT;        // (B,200)
  float* outN = outC + (size_t)BB * TT;        // (B,)

  int total = BB * NN;
  k_decode_mask<<<(total + 255) / 256, 256, 0, stream>>>(
      deltas, labels, priors, scoresT, boxesF, bh1, bh2);
  k_nms<<<BB * CC, NMS_THREADS, 0, stream>>>(scoresT, bh1, bh2, candS, candI);
  k_topk<<<BB, 256, 0, stream>>>(candS, candI, boxesF, outB, outS, outC, outN);
}